// GlowTTSMultiHeadAttentionV1_12790412607889
// MI455X (gfx1250) — compile-verified
//
#include <hip/hip_runtime.h>
#include <hip/hip_bf16.h>

// ---------------------------------------------------------------------------
// GlowTTS multi-head attention, fp32, CDNA5 (gfx1250, wave32, WMMA).
// D=512, H=4, K=128, T=1024, B=4, rel window=4, band=256.
// ---------------------------------------------------------------------------

typedef float v2f __attribute__((ext_vector_type(2)));
typedef float v8f __attribute__((ext_vector_type(8)));

#define DIM     512
#define NHEAD   4
#define KCH     128
#define SEQT    1024
#define BATCH   4
#define BAND    256
#define SW      (2 * BAND + 16)   // 528: score window per 16-query tile
#define SROW    544               // padded LDS row stride (floats)
#define NEGV    (-10000.0f)

static __device__ __forceinline__ v8f wmma_f32(v2f a, v2f b, v8f c) {
    // D(16x16,f32) = A(16x4,f32) * B(4x16,f32) + C
    return __builtin_amdgcn_wmma_f32_16x16x4_f32(
        /*neg_a=*/false, a, /*neg_b=*/false, b,
        /*c_mod=*/(short)0, c, /*reuse_a=*/false, /*reuse_b=*/false);
}

static __device__ __forceinline__ int iclamp(int x, int lo, int hi) {
    return x < lo ? lo : (x > hi ? hi : x);
}

// ---------------------------------------------------------------------------
// Generic linear layer: C[M,N] = A[M,512] * W[N,512]^T + bias[N]
// headsplit=1 : write to q/k/v layout [B,H,T,K]   (n -> h*K+kc)
// headsplit=0 : write flat [M, 512]
// Block tile 64(M) x 128(N), k staged 32 at a time in LDS.
// 8 waves: 4(M) x 2(N); each wave owns 16x64 (4 f32 WMMA accumulators).
// ---------------------------------------------------------------------------
#define GT_M 64
#define GT_N 128
#define GT_K 32

__global__ __launch_bounds__(256)
void linear_wmma_kernel(const float* __restrict__ A,
                        const float* __restrict__ W,
                        const float* __restrict__ bias,
                        float* __restrict__ out,
                        int headsplit)
{
    __shared__ float As[GT_M * GT_K];   // 8 KB
    __shared__ float Ws[GT_N * GT_K];   // 16 KB

    const int tid  = threadIdx.x;
    const int lane = tid & 31;
    const int wv   = tid >> 5;
    const int m0   = blockIdx.y * GT_M;
    const int n0   = blockIdx.x * GT_N;

    const int mrow0 = (wv & 3) * 16;     // wave M offset in block tile
    const int ncol0 = (wv >> 2) * 64;    // wave N offset in block tile

    const int hl   = lane & 15;          // half-lane id
    const int koff = (lane >> 4) * 2;    // K sub-offset per A/B layout

    v8f acc[4] = {};

    for (int k0 = 0; k0 < DIM; k0 += GT_K) {
        // Stage A tile (64x32) and W tile (128x32), float4 vectorized.
        for (int i = tid; i < (GT_M * GT_K) / 4; i += 256) {
            const int r = i >> 3, kq = (i & 7) * 4;
            *reinterpret_cast<float4*>(&As[r * GT_K + kq]) =
                *reinterpret_cast<const float4*>(A + (size_t)(m0 + r) * DIM + k0 + kq);
        }
        for (int i = tid; i < (GT_N * GT_K) / 4; i += 256) {
            const int r = i >> 3, kq = (i & 7) * 4;
            *reinterpret_cast<float4*>(&Ws[r * GT_K + kq]) =
                *reinterpret_cast<const float4*>(W + (size_t)(n0 + r) * DIM + k0 + kq);
        }
        // Prefetch next k-chunk into cache hierarchy (global_prefetch_b8).
        if (k0 + GT_K < DIM) {
            __builtin_prefetch(A + (size_t)(m0 + (tid >> 2)) * DIM + k0 + GT_K, 0, 3);
            __builtin_prefetch(W + (size_t)(n0 + (tid >> 1)) * DIM + k0 + GT_K, 0, 3);
        }
        __syncthreads();

#pragma unroll
        for (int kk = 0; kk < GT_K; kk += 4) {
            v2f a;
            a.x = As[(mrow0 + hl) * GT_K + kk + koff];
            a.y = As[(mrow0 + hl) * GT_K + kk + koff + 1];
#pragma unroll
            for (int j = 0; j < 4; ++j) {
                v2f b;
                b.x = Ws[(ncol0 + j * 16 + hl) * GT_K + kk + koff];
                b.y = Ws[(ncol0 + j * 16 + hl) * GT_K + kk + koff + 1];
                acc[j] = wmma_f32(a, b, acc[j]);
            }
        }
        __syncthreads();
    }

    // C/D layout: VGPR r -> M = r + 8*(lane>=16), N = lane&15.
    const int Mh = (lane >> 4) * 8;
#pragma unroll
    for (int j = 0; j < 4; ++j) {
#pragma unroll
        for (int r = 0; r < 8; ++r) {
            const int gm = m0 + mrow0 + r + Mh;
            const int gn = n0 + ncol0 + j * 16 + hl;
            const float v = acc[j][r] + bias[gn];
            if (headsplit) {
                const int b = gm >> 10, t = gm & (SEQT - 1);
                const int h = gn >> 7,  kc = gn & (KCH - 1);
                out[(((size_t)(b * NHEAD + h) * SEQT) + t) * KCH + kc] = v;
            } else {
                out[(size_t)gm * DIM + gn] = v;
            }
        }
    }
}

// ---------------------------------------------------------------------------
// Fused banded attention for one (b,h) and one 16-query tile.
// Scores for the full +-256 band (528 cols) live in LDS; online everything:
//   QK^T (WMMA) + rel-k logits (WMMA) + proximal bias + masks ->
//   in-wave softmax (shfl_xor, wave32) -> PV (WMMA, A from LDS probs)
//   + rel-v diagonal contribution. Output in merged-head [B,T,D] layout.
// ---------------------------------------------------------------------------
__global__ __launch_bounds__(256)
void banded_attn_kernel(const float* __restrict__ q,
                        const float* __restrict__ k,
                        const float* __restrict__ v,
                        const float* __restrict__ attn_mask,
                        const float* __restrict__ relk,
                        const float* __restrict__ relv,
                        float* __restrict__ o)
{
    __shared__ float Qs[16 * KCH];     // 8 KB   query tile
    __shared__ float Sc[16 * SROW];    // 34 KB  banded scores / probabilities
    __shared__ float Rl[16 * 16];      // rel-key logits q . relk[j]
    __shared__ float Rk[9 * KCH];      // rel key table
    __shared__ float Rv[9 * KCH];      // rel value table

    const int tid  = threadIdx.x;
    const int lane = tid & 31;
    const int wv   = tid >> 5;
    const int qt   = blockIdx.x;            // 0..63 query tile
    const int bh   = blockIdx.y;            // 0..15 (b*H + h)
    const int b    = bh >> 2, h = bh & 3;
    const int t0   = qt * 16;
    const int sbase = t0 - BAND;            // window start (may be negative)

    const float scale = 0.08838834764831845f;   // 1/sqrt(128)

    const float* qb = q + (size_t)bh * SEQT * KCH;
    const float* kb = k + (size_t)bh * SEQT * KCH;
    const float* vb = v + (size_t)bh * SEQT * KCH;

    // Stage Q tile (float4) + rel tables.
    for (int i = tid; i < 16 * KCH / 4; i += 256) {
        const int r = i >> 5, c = (i & 31) * 4;
        *reinterpret_cast<float4*>(&Qs[r * KCH + c]) =
            *reinterpret_cast<const float4*>(qb + (size_t)(t0 + r) * KCH + c);
    }
    for (int i = tid; i < 9 * KCH; i += 256) { Rk[i] = relk[i]; Rv[i] = relv[i]; }
    __syncthreads();

    const int hl   = lane & 15;
    const int koff = (lane >> 4) * 2;
    const int Mh   = (lane >> 4) * 8;

    // Wave 0: rel-key logits Rl[m][j] = sum_k Q[m][k] * relk[j][k] (j<9).
    if (wv == 0) {
        v8f accr = {};
        for (int kk = 0; kk < KCH; kk += 4) {
            v2f a;
            a.x = Qs[hl * KCH + kk + koff];
            a.y = Qs[hl * KCH + kk + koff + 1];
            v2f bb;
            bb.x = (hl < 9) ? Rk[hl * KCH + kk + koff]     : 0.0f;
            bb.y = (hl < 9) ? Rk[hl * KCH + kk + koff + 1] : 0.0f;
            accr = wmma_f32(a, bb, accr);
        }
#pragma unroll
        for (int r = 0; r < 8; ++r) Rl[(r + Mh) * 16 + hl] = accr[r];
    }
    __syncthreads();

    // Banded score tiles: 33 s-tiles of 16 striped over 8 waves.
    for (int st = wv; st < SW / 16; st += 8) {
        const int s0 = st * 16;
        v8f acc = {};
        for (int kk = 0; kk < KCH; kk += 4) {
            v2f a;
            a.x = Qs[hl * KCH + kk + koff];
            a.y = Qs[hl * KCH + kk + koff + 1];
            const int s  = sbase + s0 + hl;
            const int sc = iclamp(s, 0, SEQT - 1);
            const float* kp = kb + (size_t)sc * KCH + kk + koff;
            v2f bb; bb.x = kp[0]; bb.y = kp[1];
            acc = wmma_f32(a, bb, acc);
        }
#pragma unroll
        for (int r = 0; r < 8; ++r) {
            const int M = r + Mh;
            const int t = t0 + M;
            const int s = sbase + s0 + hl;
            const int d = s - t;
            float val;
            const bool inband = (s >= 0) && (s < SEQT) && (d >= -BAND) && (d <= BAND);
            if (!inband) {
                val = NEGV;
            } else if (attn_mask[b * SEQT + t] * attn_mask[b * SEQT + s] == 0.0f) {
                val = NEGV;
            } else {
                val = acc[r] * scale;
                if (d >= -4 && d <= 4) val += Rl[M * 16 + (d + 4)] * scale;
                val -= __logf(1.0f + fabsf((float)d));   // proximal bias
            }
            Sc[M * SROW + s0 + hl] = val;
        }
    }
    __syncthreads();

    // Softmax: wave wv owns rows 2wv (lanes 0-15) and 2wv+1 (lanes 16-31).
    {
        const int row = 2 * wv + (lane >> 4);
        float m = -3.4e38f;
        for (int c = hl; c < SW; c += 16) m = fmaxf(m, Sc[row * SROW + c]);
#pragma unroll
        for (int off = 8; off >= 1; off >>= 1) m = fmaxf(m, __shfl_xor(m, off, 32));
        float ssum = 0.0f;
        for (int c = hl; c < SW; c += 16) {
            const float e = __expf(Sc[row * SROW + c] - m);
            Sc[row * SROW + c] = e;
            ssum += e;
        }
#pragma unroll
        for (int off = 8; off >= 1; off >>= 1) ssum += __shfl_xor(ssum, off, 32);
        const float inv = 1.0f / ssum;
        for (int c = hl; c < SW; c += 16) Sc[row * SROW + c] *= inv;
    }
    __syncthreads();

    // PV: wave wv computes head-dim chunk [16*wv, 16*wv+16).
    const int kc0 = wv * 16;
    v8f acc = {};
    for (int kk = 0; kk < SW; kk += 4) {
        v2f a;
        a.x = Sc[hl * SROW + kk + koff];
        a.y = Sc[hl * SROW + kk + koff + 1];
        const int s1 = iclamp(sbase + kk + koff, 0, SEQT - 1);
        const int s2 = iclamp(sbase + kk + koff + 1, 0, SEQT - 1);
        v2f bb;
        bb.x = vb[(size_t)s1 * KCH + kc0 + hl];
        bb.y = vb[(size_t)s2 * KCH + kc0 + hl];
        acc = wmma_f32(a, bb, acc);
    }
#pragma unroll
    for (int r = 0; r < 8; ++r) {
        const int M = r + Mh;
        const int t = t0 + M;
        float ov = acc[r];
#pragma unroll
        for (int dd = -4; dd <= 4; ++dd) {
            const int s = t + dd;
            if (s >= 0 && s < SEQT)
                ov += Sc[M * SROW + (s - sbase)] * Rv[(dd + 4) * KCH + kc0 + hl];
        }
        // merged-head output layout [B, T, H*K]
        o[(((size_t)b * SEQT + t) * NHEAD + h) * KCH + kc0 + hl] = ov;
    }
}

// ---------------------------------------------------------------------------
extern "C" void kernel_launch(void* const* d_in, const int* in_sizes, int n_in,
                              void* d_out, int out_size, void* d_ws, size_t ws_size,
                              hipStream_t stream) {
    const float* x    = (const float*)d_in[0];
    const float* c    = (const float*)d_in[1];
    const float* mask = (const float*)d_in[2];
    const float* Wq   = (const float*)d_in[3];
    const float* bq   = (const float*)d_in[4];
    const float* Wk   = (const float*)d_in[5];
    const float* bk   = (const float*)d_in[6];
    const float* Wv   = (const float*)d_in[7];
    const float* bv   = (const float*)d_in[8];
    const float* Wo   = (const float*)d_in[9];
    const float* bo   = (const float*)d_in[10];
    const float* relk = (const float*)d_in[11];
    const float* relv = (const float*)d_in[12];
    float* out = (float*)d_out;

    const size_t NQ = (size_t)BATCH * NHEAD * SEQT * KCH;   // 2,097,152 floats
    float* ws = (float*)d_ws;
    float* qbuf = ws;
    float* kbuf = ws + NQ;
    float* vbuf = ws + 2 * NQ;
    float* obuf = ws + 3 * NQ;

    const dim3 ggrid(DIM / GT_N, (BATCH * SEQT) / GT_M);    // (4, 64)
    const dim3 gblk(256);

    linear_wmma_kernel<<<ggrid, gblk, 0, stream>>>(x, Wq, bq, qbuf, 1);
    linear_wmma_kernel<<<ggrid, gblk, 0, stream>>>(c, Wk, bk, kbuf, 1);
    linear_wmma_kernel<<<ggrid, gblk, 0, stream>>>(c, Wv, bv, vbuf, 1);

    banded_attn_kernel<<<dim3(SEQT / 16, BATCH * NHEAD), gblk, 0, stream>>>(
        qbuf, kbuf, vbuf, mask, relk, relv, obuf);

    linear_wmma_kernel<<<ggrid, gblk, 0, stream>>>(obuf, Wo, bo, out, 0);
}